// SeriesDecompDEMA_25211458028031
// MI455X (gfx1250) — compile-verified
//
#include <hip/hip_runtime.h>

// Holt double-exponential smoothing (DEMA decomposition)
//   x: (32, 4096, 256) f32;  out = [res | ma], each 32*4096*256 f32
// Per-series linear recurrence h_t = M h_{t-1} + v x_t rewritten as blocked
// 16x16x16 f32 matmuls (V_WMMA_F32_16X16X4_F32 chained x4) with a 2-scalar
// cross-block state carried via lane shuffles.

#define ALPHA 0.2f
#define BETA  0.1f

typedef __attribute__((ext_vector_type(2))) float v2f;
typedef __attribute__((ext_vector_type(8))) float v8f;

struct Coef {
  float c[16];    // (M^j v)_s  -> Toeplitz coeffs of W
  float d[16];    // (M^j v)_b  -> b-state input coeffs
  float p00[16];  // (M^{i+1})_{00}
  float p01[16];  // (M^{i+1})_{01}
  float q10, q11; // (M^16) row 1
};

static constexpr Coef make_coef() {
  Coef k{};
  const double A = 0.2, Bt = 0.1;
  const double M00 = 1.0 - A, M01 = 1.0 - A;
  const double M10 = -A * Bt, M11 = 1.0 - A * Bt;
  const double v0 = A, v1 = A * Bt;
  double m00 = 1, m01 = 0, m10 = 0, m11 = 1; // M^0
  for (int j = 0; j < 16; ++j) {
    k.c[j] = (float)(m00 * v0 + m01 * v1);
    k.d[j] = (float)(m10 * v0 + m11 * v1);
    const double n00 = M00 * m00 + M01 * m10;
    const double n01 = M00 * m01 + M01 * m11;
    const double n10 = M10 * m00 + M11 * m10;
    const double n11 = M10 * m01 + M11 * m11;
    m00 = n00; m01 = n01; m10 = n10; m11 = n11; // now M^{j+1}
    k.p00[j] = (float)m00;
    k.p01[j] = (float)m01;
  }
  k.q10 = (float)m10; // M^16
  k.q11 = (float)m11;
  return k;
}

__constant__ Coef CO = make_coef();

__global__ __launch_bounds__(256)
void dema_wmma_kernel(const float* __restrict__ x,
                      float* __restrict__ res,
                      float* __restrict__ ma) {
  constexpr int T = 4096;
  constexpr int C = 256;

  const int lane = threadIdx.x & 31;
  const int wave = blockIdx.x * (blockDim.x >> 5) + (threadIdx.x >> 5); // 0..511
  const int batch = wave >> 4;          // 0..31
  const int c0 = (wave & 15) << 4;      // channel group base
  const int N  = lane & 15;             // column (channel) / A-row index
  const int hi = lane >> 4;             // half-wave selector

  const size_t base = (size_t)batch * T * C + c0;
  const float* __restrict__ xb  = x   + base;
  float* __restrict__ resb      = res + base;
  float* __restrict__ mab       = ma  + base;

  // ---- per-lane constant fragments (computed once) ----
  // A = W (16x16 lower-tri Toeplitz), split into four 16x4 K-blocks.
  // f32 A layout: lanes0-15 VGPR0=K0,VGPR1=K1 ; lanes16-31 VGPR0=K2,VGPR1=K3
  v2f afrag[4];
  float dd0[4], dd1[4];
#pragma unroll
  for (int kb = 0; kb < 4; ++kb) {
    const int k0 = 4 * kb + 2 * hi;
    afrag[kb].x = (k0     <= N) ? CO.c[N - k0]     : 0.0f;
    afrag[kb].y = (k0 + 1 <= N) ? CO.c[N - k0 - 1] : 0.0f;
    dd0[kb] = CO.d[15 - k0];
    dd1[kb] = CO.d[14 - k0];
  }
  float p0r[8], p1r[8];
#pragma unroll
  for (int g = 0; g < 8; ++g) {
    p0r[g] = CO.p00[g + 8 * hi];
    p1r[g] = CO.p01[g + 8 * hi];
  }
  const float q10 = CO.q10, q11 = CO.q11;

  // ---- block 0 (t = 0..15) scalar: seeds the recurrence ----
  float s  = xb[N];         // s0 = x[b,0,c]; ma[0] = s0; res[0] = 0
  float x1 = xb[C + N];
  float bt = x1 - s;        // b0
  if (hi == 0) { mab[N] = s; resb[N] = 0.0f; }
  {
    const float sn = ALPHA * x1 + (1.0f - ALPHA) * (s + bt);
    bt = BETA * (sn - s) + (1.0f - BETA) * bt;
    s = sn;
    if (hi == 0) { mab[C + N] = s; resb[C + N] = x1 - s; }
  }
#pragma unroll
  for (int t = 2; t < 16; ++t) {
    const float xt = xb[(size_t)t * C + N];
    const float sn = ALPHA * xt + (1.0f - ALPHA) * (s + bt);
    bt = BETA * (sn - s) + (1.0f - BETA) * bt;
    s = sn;
    if (hi == 0) { mab[(size_t)t * C + N] = s; resb[(size_t)t * C + N] = xt - s; }
  }
  float s_in = s, b_in = bt;

  // ---- B-fragment double buffer: block j=1 ----
  v2f bf[4], bfn[4];
  {
    const float* p = xb + (size_t)16 * C;
#pragma unroll
    for (int kb = 0; kb < 4; ++kb) {
      const int r0 = 4 * kb + 2 * hi;
      bf[kb].x = p[(size_t)r0 * C + N];
      bf[kb].y = p[(size_t)(r0 + 1) * C + N];
    }
  }

  // ---- main loop: 255 blocks of 16 timesteps via WMMA ----
  for (int j = 1; j < 256; ++j) {
    const int t0 = j << 4;
    const float* pcur = xb + (size_t)t0 * C;

    if (j < 255) {
      const float* pn = pcur + (size_t)16 * C;
#pragma unroll
      for (int kb = 0; kb < 4; ++kb) {
        const int r0 = 4 * kb + 2 * hi;
        bfn[kb].x = pn[(size_t)r0 * C + N];
        bfn[kb].y = pn[(size_t)(r0 + 1) * C + N];
      }
      __builtin_prefetch(pcur + (size_t)32 * C + N, 0, 0); // global_prefetch_b8
    }

    // C seeded with state contribution: s_i gets (M^{i+1} h_in)_s
    v8f acc;
#pragma unroll
    for (int g = 0; g < 8; ++g) acc[g] = p0r[g] * s_in + p1r[g] * b_in;

#if __has_builtin(__builtin_amdgcn_wmma_f32_16x16x4_f32)
    acc = __builtin_amdgcn_wmma_f32_16x16x4_f32(false, afrag[0], false, bf[0],
                                                (short)0, acc, false, false);
    acc = __builtin_amdgcn_wmma_f32_16x16x4_f32(false, afrag[1], false, bf[1],
                                                (short)0, acc, false, false);
    acc = __builtin_amdgcn_wmma_f32_16x16x4_f32(false, afrag[2], false, bf[2],
                                                (short)0, acc, false, false);
    acc = __builtin_amdgcn_wmma_f32_16x16x4_f32(false, afrag[3], false, bf[3],
                                                (short)0, acc, false, false);
#else
    // Scalar fallback: full 16-step scan per lane, then pick D-layout rows.
    {
      float ss = s_in, bb = b_in, sv[16];
#pragma unroll
      for (int i = 0; i < 16; ++i) {
        const float xt = pcur[(size_t)i * C + N];
        const float sn = ALPHA * xt + (1.0f - ALPHA) * (ss + bb);
        bb = BETA * (sn - ss) + (1.0f - BETA) * bb;
        ss = sn;
        sv[i] = ss;
      }
#pragma unroll
      for (int g = 0; g < 8; ++g) acc[g] = sv[g + 8 * hi];
    }
#endif

    // cross-block state update (off the WMMA critical path except s15)
    float part = 0.0f;
#pragma unroll
    for (int kb = 0; kb < 4; ++kb)
      part += dd0[kb] * bf[kb].x + dd1[kb] * bf[kb].y;
    part += __shfl_xor(part, 16, 32);            // combine half-wave partials
    const float s15 = __shfl(acc[7], 16 + N, 32); // M=15 row lives in lanes>=16
    b_in = q10 * s_in + q11 * b_in + part;
    s_in = s15;

    // stores (D layout): ma = acc ; res = x - acc (reload hits L0/WGP$)
    float* mo = mab  + (size_t)t0 * C;
    float* ro = resb + (size_t)t0 * C;
#pragma unroll
    for (int g = 0; g < 8; ++g) {
      const int tr = g + 8 * hi;
      const float xv = pcur[(size_t)tr * C + N];
      mo[(size_t)tr * C + N] = acc[g];
      ro[(size_t)tr * C + N] = xv - acc[g];
    }

#pragma unroll
    for (int kb = 0; kb < 4; ++kb) bf[kb] = bfn[kb];
  }
}

extern "C" void kernel_launch(void* const* d_in, const int* in_sizes, int n_in,
                              void* d_out, int out_size, void* d_ws, size_t ws_size,
                              hipStream_t stream) {
  (void)in_sizes; (void)n_in; (void)out_size; (void)d_ws; (void)ws_size;
  const float* x = (const float*)d_in[0];
  float* out = (float*)d_out;
  const size_t n = (size_t)32 * 4096 * 256;
  float* res = out;       // first output
  float* ma  = out + n;   // second output
  // 8192 series / 16 per wave = 512 waves = 64 blocks x 8 waves(256 thr)
  dim3 grid(64), block(256);
  hipLaunchKernelGGL(dema_wmma_kernel, grid, block, 0, stream, x, res, ma);
}